// CriticalLoss_90117003804742
// MI455X (gfx1250) — compile-verified
//
#include <hip/hip_runtime.h>
#include <hip/hip_bf16.h>
#include <math.h>

typedef __attribute__((ext_vector_type(2))) float v2f;
typedef __attribute__((ext_vector_type(8))) float v8f;

#define N_ART   10
#define NSAMP   50
#define SPAD    64          // 50 padded to 4 tiles of 16
#define NPAIR   (N_ART * N_ART)
#define WAVES_PER_BLOCK 8

// Pass 1: one block per frame (BT = bs*seq = 512 blocks, 256 threads = 8 wave32).
// Computes sum_{i,j} mask[b,i,j] * min-pairwise-distance(i,j) -> partial[b].
__global__ void __launch_bounds__(256)
critloss_frame_kernel(const float* __restrict__ outputs,
                      const float* __restrict__ masks,
                      float* __restrict__ partial) {
    __shared__ float sx [N_ART * SPAD];
    __shared__ float sy [N_ART * SPAD];
    __shared__ float ssq[N_ART * SPAD];
    __shared__ float wsum[WAVES_PER_BLOCK];

    const int b   = blockIdx.x;
    const int tid = threadIdx.x;
    // outputs frame layout: (n_art, 2, n_samples) contiguous per frame
    const float* fr = outputs + (size_t)b * (N_ART * 2 * NSAMP);

    // Stage padded points + squared norms into LDS.
    for (int idx = tid; idx < N_ART * SPAD; idx += blockDim.x) {
        const int a = idx >> 6;          // / SPAD
        const int s = idx & (SPAD - 1);  // % SPAD
        float x, y;
        if (s < NSAMP) {
            x = fr[a * (2 * NSAMP) + s];
            y = fr[a * (2 * NSAMP) + NSAMP + s];
        } else {
            // Per-articulator sentinel: cross-articulator pad distances are huge;
            // same-articulator pad collisions only affect i==j pairs (true min = 0).
            x = 1.0e6f * (float)(a + 1);
            y = x;
        }
        sx[idx]  = x;
        sy[idx]  = y;
        ssq[idx] = x * x + y * y;
    }
    __syncthreads();

    const int  wid    = tid >> 5;
    const int  lane   = tid & 31;
    const int  lane16 = lane & 15;
    const bool lo     = (lane < 16);
    const int  hi8    = lo ? 0 : 8;      // C/D VGPR r holds M = r + hi8

    float acc = 0.0f;                    // per-wave masked-distance accumulator (lane 0)

    // Each wave handles pairs p = wid, wid+8, ... (wave-uniform loop: EXEC stays full
    // for WMMA as required).
    for (int p = wid; p < NPAIR; p += WAVES_PER_BLOCK) {
        const int i = p / N_ART;
        const int j = p - i * N_ART;

        // A fragment (16x4 f32): lanes 0-15 hold row M=lane, v0=K0(x), v1=K1(y);
        // lanes 16-31 hold K=2,3 which are zero padding. B (4x16) symmetric.
        // Loads are unconditional (address in-range for every lane); zeroing of
        // the upper half-wave is a v_cndmask select, keeping EXEC untouched and
        // avoiding predicated ds_loads.
        v2f afrag[4], bfrag[4];
        float sqj[4];
#pragma unroll
        for (int t = 0; t < 4; ++t) {
            const int r = t * 16 + lane16;
            const float ax = sx[i * SPAD + r];
            const float ay = sy[i * SPAD + r];
            const float bx = sx[j * SPAD + r];
            const float by = sy[j * SPAD + r];
            v2f a, bb;
            a.x  = lo ? ax : 0.0f;
            a.y  = lo ? ay : 0.0f;
            bb.x = lo ? bx : 0.0f;
            bb.y = lo ? by : 0.0f;
            afrag[t] = a;
            bfrag[t] = bb;
            sqj[t] = ssq[j * SPAD + r];  // per-lane column norm, N = lane16
        }

        float minv = 3.0e38f;
#pragma unroll
        for (int rt = 0; rt < 4; ++rt) {
            float sqi[8];
#pragma unroll
            for (int r = 0; r < 8; ++r)
                sqi[r] = ssq[i * SPAD + rt * 16 + hi8 + r];
#pragma unroll
            for (int ct = 0; ct < 4; ++ct) {
                v8f c = {};
                // dot tile: D = A(16x4) * B(4x16), f32 accumulate
                c = __builtin_amdgcn_wmma_f32_16x16x4_f32(
                        /*neg_a=*/false, afrag[rt],
                        /*neg_b=*/false, bfrag[ct],
                        /*c_mod=*/(short)0, c,
                        /*reuse_a=*/false, /*reuse_b=*/false);
#pragma unroll
                for (int r = 0; r < 8; ++r) {
                    const float d2 = sqi[r] + sqj[ct] - 2.0f * c[r];
                    minv = fminf(minv, d2);
                }
            }
        }

        // wave32 min-reduction
#pragma unroll
        for (int off = 16; off >= 1; off >>= 1)
            minv = fminf(minv, __shfl_xor(minv, off, 32));

        const float d = (minv > 0.0f) ? sqrtf(minv) : 0.0f;
        if (lane == 0)
            acc += masks[(size_t)b * NPAIR + p] * d;
    }

    if (lane == 0) wsum[wid] = acc;
    __syncthreads();
    if (tid == 0) {
        float s = 0.0f;
#pragma unroll
        for (int w = 0; w < WAVES_PER_BLOCK; ++w) s += wsum[w];
        partial[b] = s;
    }
}

// Pass 2: deterministic tree reduction of BT partials -> scalar mean.
__global__ void __launch_bounds__(256)
critloss_reduce_kernel(const float* __restrict__ partial, float* __restrict__ out,
                       int n, float scale) {
    __shared__ float buf[256];
    const int tid = threadIdx.x;
    float s = 0.0f;
    for (int idx = tid; idx < n; idx += 256) s += partial[idx];
    buf[tid] = s;
    __syncthreads();
#pragma unroll
    for (int off = 128; off >= 1; off >>= 1) {
        if (tid < off) buf[tid] += buf[tid + off];
        __syncthreads();
    }
    if (tid == 0) out[0] = buf[0] * scale;
}

extern "C" void kernel_launch(void* const* d_in, const int* in_sizes, int n_in,
                              void* d_out, int out_size, void* d_ws, size_t ws_size,
                              hipStream_t stream) {
    (void)n_in; (void)out_size; (void)ws_size;
    const float* outputs = (const float*)d_in[0];   // (4,128,10,2,50) f32
    const float* masks   = (const float*)d_in[1];   // (4,128,10,10)  f32
    float* out = (float*)d_out;
    float* partial = (float*)d_ws;                  // BT floats of scratch

    const int bt = in_sizes[1] / NPAIR;             // 512 frames
    const float scale = 1.0f / (float)(bt * NPAIR); // mean over masks*min_dists

    critloss_frame_kernel<<<bt, 256, 0, stream>>>(outputs, masks, partial);
    critloss_reduce_kernel<<<1, 256, 0, stream>>>(partial, out, bt, scale);
}